// FlashMQAttention_90787018703129
// MI455X (gfx1250) — compile-verified
//
#include <hip/hip_runtime.h>

typedef __attribute__((ext_vector_type(2))) float v2f;
typedef __attribute__((ext_vector_type(8))) float v8f;

#define B_      64
#define S_      4096
#define HID_    2048
#define NH_     16
#define HD_     128
#define NKV_    2304                 // HID + 2*HD
#define SPLITS_ 8
#define SPLIT_LEN_ (S_ / SPLITS_)    // 512
#define SCALE_  0.08838834764831845f // HD^-0.5

// ---------------------------------------------------------------------------
// C[M,N] = A[M,K] @ B[K,N] + bias[N], fp32, one wave per 16x16 C tile.
// Both operands staged in LDS: A with row stride 132 (conflict-free strided
// fragment reads), B with row stride 16 (fragment reads pair into 2addr).
// ---------------------------------------------------------------------------
__global__ __launch_bounds__(32)
void gemm_bias_wmma(const float* __restrict__ A, const float* __restrict__ Bm,
                    const float* __restrict__ bias, float* __restrict__ C,
                    int M, int N, int K) {
  __shared__ float As[16 * 132];
  __shared__ float Bs[128 * 16];
  const int lane = threadIdx.x;
  const int tn = blockIdx.x * 16;
  const int tm = blockIdx.y * 16;
  const int rr = lane & 15;
  const int off = ((lane >> 4) & 1) * 2;   // K sub-offset for upper half-wave

  v8f acc = {};
  for (int kb = 0; kb < K; kb += 128) {
    __syncthreads();
    // stage A[tm..tm+16, kb..kb+128] -> LDS, coalesced float4
    #pragma unroll
    for (int it = 0; it < 16; ++it) {
      const int i = it * 32 + lane;
      const int row = i >> 5;
      const int d4 = (i & 31) << 2;
      const float4 v = *reinterpret_cast<const float4*>(
          &A[(size_t)(tm + row) * K + kb + d4]);
      float* dst = &As[row * 132 + d4];
      dst[0] = v.x; dst[1] = v.y; dst[2] = v.z; dst[3] = v.w;
    }
    // stage B[kb..kb+128, tn..tn+16] -> LDS
    #pragma unroll
    for (int it = 0; it < 16; ++it) {
      const int i = it * 32 + lane;
      const int row = i >> 2;
      const int c4 = (i & 3) << 2;
      const float4 v = *reinterpret_cast<const float4*>(
          &Bm[(size_t)(kb + row) * N + tn + c4]);
      float* dst = &Bs[row * 16 + c4];
      dst[0] = v.x; dst[1] = v.y; dst[2] = v.z; dst[3] = v.w;
    }
    __syncthreads();
    #pragma unroll 8
    for (int k2 = 0; k2 < 128; k2 += 4) {
      v2f a, b;
      a[0] = As[rr * 132 + k2 + off];
      a[1] = As[rr * 132 + k2 + off + 1];
      b[0] = Bs[(k2 + off) * 16 + rr];
      b[1] = Bs[(k2 + off + 1) * 16 + rr];
      acc = __builtin_amdgcn_wmma_f32_16x16x4_f32(
          false, a, false, b, (short)0, acc, false, false);
    }
  }
  const float bz = bias[tn + rr];
  #pragma unroll
  for (int r = 0; r < 8; ++r) {
    const int row = r + ((lane >> 4) << 3);
    C[(size_t)(tm + row) * N + tn + rr] = acc[r] + bz;
  }
}

// ---------------------------------------------------------------------------
// CDNA5 async copy: one 16-position KV tile (contiguous [s][k|v][d] 16KB
// block) -> padded LDS K/V buffers, 32 global_load_async_to_lds_b128
// (each lane carries 16B; per instruction lanes cover one 512B K- or V-row).
// Tracked by ASYNCcnt; caller synchronizes with s_wait_asynccnt.
// ---------------------------------------------------------------------------
__device__ __forceinline__ void issue_tile_async(const float* __restrict__ kvp,
                                                 float* kbuf, float* vbuf,
                                                 int lane) {
  #pragma unroll
  for (int it = 0; it < 32; ++it) {
    const float* g = kvp + it * 128 + (lane << 2);
    float* dst = ((it & 1) ? vbuf : kbuf) + (it >> 1) * 132 + (lane << 2);
    const unsigned lds_off = (unsigned)(unsigned long long)dst;  // addr[31:0]
    asm volatile("global_load_async_to_lds_b128 %0, %1, off"
                 :: "v"(lds_off), "v"(g) : "memory");
  }
}

// ---------------------------------------------------------------------------
// Flash-decode MQA partial: one wave per (sequence, split of 512 KV slots).
// All 16 heads ride the M=16 axis of the WMMA tiles (MQA: shared K/V).
// Double-buffered async KV tiles: fetch of tile t+1 overlaps compute of t.
// ---------------------------------------------------------------------------
__global__ __launch_bounds__(32)
void mqa_partial(const float* __restrict__ qkv,
                 const float* __restrict__ layer_past,
                 float* __restrict__ o_part,
                 float* __restrict__ m_part,
                 float* __restrict__ l_part) {
  __shared__ float Qt[16 * 132];
  __shared__ float Kt[2][16 * 132];
  __shared__ float Vt[2][16 * 132];
  __shared__ float Pl[16 * 17];

  const int lane = threadIdx.x;
  const int sp = blockIdx.x;
  const int b  = blockIdx.y;
  const int rr = lane & 15;
  const int hi = (lane >> 4) & 1;
  const int off = hi * 2;

  // Q for this sequence: 16 heads x 128 dims -> LDS
  const float* qp = qkv + (size_t)b * NKV_;
  #pragma unroll
  for (int it = 0; it < 16; ++it) {
    const int i = it * 32 + lane;
    const int h = i >> 5;
    const int d4 = (i & 31) << 2;
    const float4 v = *reinterpret_cast<const float4*>(&qp[h * HD_ + d4]);
    float* dst = &Qt[h * 132 + d4];
    dst[0] = v.x; dst[1] = v.y; dst[2] = v.z; dst[3] = v.w;
  }

  v8f o0 = {}, o1 = {}, o2 = {}, o3 = {}, o4 = {}, o5 = {}, o6 = {}, o7 = {};
  float m_i[8], l_i[8];
  #pragma unroll
  for (int r = 0; r < 8; ++r) { m_i[r] = -__builtin_inff(); l_i[r] = 0.f; }

  const int s_begin = sp * SPLIT_LEN_;
  const int ntiles = SPLIT_LEN_ / 16;
  const float* kv_base = layer_past + ((size_t)b * S_ + s_begin) * (2 * HD_);

  // prefetch tile 0
  issue_tile_async(kv_base, Kt[0], Vt[0], lane);

  for (int t = 0; t < ntiles; ++t) {
    const int cur = t & 1;
    const int s0 = s_begin + t * 16;
    // tile t's async LDS writes complete (loads finish in order; only tile t
    // is outstanding here, tile t+1 has not been issued yet)
    asm volatile("s_wait_asynccnt 0x0" ::: "memory");
    if (t + 1 < ntiles)
      issue_tile_async(kv_base + (size_t)(t + 1) * 16 * (2 * HD_),
                       Kt[cur ^ 1], Vt[cur ^ 1], lane);
    // current token: its k/v live in the qkv buffer, not the (const) cache
    if (s0 + 16 > S_ - 1) {
      const int ls = (S_ - 1) - s0;
      for (int d = lane; d < HD_; d += 32) {
        Kt[cur][ls * 132 + d] = qp[HID_ + d];
        Vt[cur][ls * 132 + d] = qp[HID_ + HD_ + d];
      }
    }
    __syncthreads();

    // scores[16 heads x 16 pos] = Q (16x128) @ K^T (128x16)
    const float* Kc = Kt[cur];
    const float* Vc = Vt[cur];
    v8f sc = {};
    #pragma unroll 8
    for (int d = 0; d < HD_; d += 4) {
      v2f a, bv;
      a[0]  = Qt[rr * 132 + d + off];
      a[1]  = Qt[rr * 132 + d + off + 1];
      bv[0] = Kc[rr * 132 + d + off];
      bv[1] = Kc[rr * 132 + d + off + 1];
      sc = __builtin_amdgcn_wmma_f32_16x16x4_f32(
          false, a, false, bv, (short)0, sc, false, false);
    }

    // online softmax: row = head, N across 16-lane halves -> butterfly shuffles
    #pragma unroll
    for (int r = 0; r < 8; ++r) {
      float x = sc[r] * SCALE_;
      float rmax = x;
      rmax = fmaxf(rmax, __shfl_xor(rmax, 1, 32));
      rmax = fmaxf(rmax, __shfl_xor(rmax, 2, 32));
      rmax = fmaxf(rmax, __shfl_xor(rmax, 4, 32));
      rmax = fmaxf(rmax, __shfl_xor(rmax, 8, 32));
      const float mn = fmaxf(m_i[r], rmax);
      const float corr = __expf(m_i[r] - mn);
      const float p = __expf(x - mn);
      float rs = p;
      rs += __shfl_xor(rs, 1, 32);
      rs += __shfl_xor(rs, 2, 32);
      rs += __shfl_xor(rs, 4, 32);
      rs += __shfl_xor(rs, 8, 32);
      l_i[r] = l_i[r] * corr + rs;
      m_i[r] = mn;
      Pl[(r + (hi << 3)) * 17 + rr] = p;
      o0[r] *= corr; o1[r] *= corr; o2[r] *= corr; o3[r] *= corr;
      o4[r] *= corr; o5[r] *= corr; o6[r] *= corr; o7[r] *= corr;
    }
    __syncthreads();

    // O(16x128) += P(16x16) @ V(16x128): 4 K-steps x 8 N-tiles
    #pragma unroll
    for (int s2 = 0; s2 < 16; s2 += 4) {
      v2f a;
      a[0] = Pl[rr * 17 + s2 + off];
      a[1] = Pl[rr * 17 + s2 + off + 1];
      const float* v0 = &Vc[(s2 + off) * 132 + rr];
      const float* v1 = &Vc[(s2 + off + 1) * 132 + rr];
      #define PV_TILE(on, n)                                              \
        { v2f bv; bv[0] = v0[(n) * 16]; bv[1] = v1[(n) * 16];             \
          on = __builtin_amdgcn_wmma_f32_16x16x4_f32(                     \
              false, a, false, bv, (short)0, on, false, false); }
      PV_TILE(o0, 0) PV_TILE(o1, 1) PV_TILE(o2, 2) PV_TILE(o3, 3)
      PV_TILE(o4, 4) PV_TILE(o5, 5) PV_TILE(o6, 6) PV_TILE(o7, 7)
      #undef PV_TILE
    }
    __syncthreads();
  }

  float* ob = o_part + (size_t)(b * SPLITS_ + sp) * NH_ * HD_;
  #pragma unroll
  for (int r = 0; r < 8; ++r) {
    const int row = r + (hi << 3);
    float* orow = ob + row * HD_ + rr;
    orow[0]  = o0[r]; orow[16] = o1[r]; orow[32]  = o2[r]; orow[48]  = o3[r];
    orow[64] = o4[r]; orow[80] = o5[r]; orow[96]  = o6[r]; orow[112] = o7[r];
  }
  if (rr == 0) {
    #pragma unroll
    for (int r = 0; r < 8; ++r) {
      const int row = r + (hi << 3);
      m_part[(b * SPLITS_ + sp) * NH_ + row] = m_i[r];
      l_part[(b * SPLITS_ + sp) * NH_ + row] = l_i[r];
    }
  }
}

// ---------------------------------------------------------------------------
// Merge the 8 split partials (flash-decode combine) -> attn [B, NH*HD]
// ---------------------------------------------------------------------------
__global__ __launch_bounds__(256)
void mqa_combine(const float* __restrict__ o_part,
                 const float* __restrict__ m_part,
                 const float* __restrict__ l_part,
                 float* __restrict__ attn) {
  const int b = blockIdx.x;
  for (int e = threadIdx.x; e < NH_ * HD_; e += 256) {
    const int h = e >> 7;
    float M = -__builtin_inff();
    #pragma unroll
    for (int sp = 0; sp < SPLITS_; ++sp)
      M = fmaxf(M, m_part[(b * SPLITS_ + sp) * NH_ + h]);
    float L = 0.f, acc = 0.f;
    #pragma unroll
    for (int sp = 0; sp < SPLITS_; ++sp) {
      const float w = __expf(m_part[(b * SPLITS_ + sp) * NH_ + h] - M);
      L += w * l_part[(b * SPLITS_ + sp) * NH_ + h];
      acc += w * o_part[((size_t)(b * SPLITS_ + sp) * NH_ + h) * HD_ + (e & 127)];
    }
    attn[(size_t)b * HID_ + e] = acc / L;
  }
}

// ---------------------------------------------------------------------------
extern "C" void kernel_launch(void* const* d_in, const int* in_sizes, int n_in,
                              void* d_out, int out_size, void* d_ws, size_t ws_size,
                              hipStream_t stream) {
  const float* hidden     = (const float*)d_in[0];
  const float* W_attn     = (const float*)d_in[1];
  const float* b_attn     = (const float*)d_in[2];
  const float* W_proj     = (const float*)d_in[3];
  const float* b_proj     = (const float*)d_in[4];
  const float* layer_past = (const float*)d_in[5];

  float* ws     = (float*)d_ws;
  float* qkv    = ws;                                   // 64*2304
  float* attn   = qkv + B_ * NKV_;                      // 64*2048
  float* o_part = attn + B_ * HID_;                     // 64*8*16*128
  float* m_part = o_part + (size_t)B_ * SPLITS_ * NH_ * HD_;
  float* l_part = m_part + B_ * SPLITS_ * NH_;

  gemm_bias_wmma<<<dim3(NKV_ / 16, B_ / 16), 32, 0, stream>>>(
      hidden, W_attn, b_attn, qkv, B_, NKV_, HID_);
  mqa_partial<<<dim3(SPLITS_, B_), 32, 0, stream>>>(
      qkv, layer_past, o_part, m_part, l_part);
  mqa_combine<<<B_, 256, 0, stream>>>(o_part, m_part, l_part, attn);
  gemm_bias_wmma<<<dim3(HID_ / 16, B_ / 16), 32, 0, stream>>>(
      attn, W_proj, b_proj, (float*)d_out, B_, HID_, HID_);
}